// SELRU_10505490006122
// MI455X (gfx1250) — compile-verified
//
#include <hip/hip_runtime.h>

#define Dd 256
#define Nn 64
#define Bb 2
#define Ll 2048
#define Mm (Bb * Ll)       // 4096 rows
#define NCH 32             // scan chunks along L
#define CLEN (Ll / NCH)    // 64 steps per chunk

typedef __attribute__((ext_vector_type(2))) float v2f;
typedef __attribute__((ext_vector_type(8))) float v8f;

// ---------------------------------------------------------------------------
// Generic f32 WMMA GEMM: C[M,Nout] = act(A[M,Kd] * W[Nout,Kd]^T + bias)
// One wave computes a 16x64 strip using V_WMMA_F32_16X16X4_F32.
// Optional column-split store (C0 gets cols < splitCol, C1 the rest).
// ---------------------------------------------------------------------------
__global__ __launch_bounds__(256) void wmma_gemm_f32(
    const float* __restrict__ A, const float* __restrict__ W,
    const float* __restrict__ bias, float* __restrict__ C0,
    float* __restrict__ C1, int splitCol, int M, int Kd, int Nout, int act)
{
  const int wave   = (blockIdx.x * blockDim.x + threadIdx.x) >> 5;
  const int lane   = threadIdx.x & 31;
  const int tilesM = M >> 4;
  const int m0 = (wave % tilesM) << 4;
  const int n0 = (wave / tilesM) << 6;
  const int half = lane >> 4;      // 0: K rows {0,1}, 1: K rows {2,3}
  const int l15  = lane & 15;
  const int koff = half << 1;

  // A fragment (16x4): lanes 0-15 rows M, VGPR0/1 = K0/K1; lanes 16-31 K2/K3
  const float* Arow = A + (size_t)(m0 + l15) * Kd + koff;
  // B fragment (4x16): mirrored layout, B[k][n] = W[n*Kd + k]
  const float* Wrow = W + (size_t)(n0 + l15) * Kd + koff;

  v8f acc[4] = {};
  #pragma unroll 2
  for (int k = 0; k < Kd; k += 4) {
    v2f a = *(const v2f*)(Arow + k);
    __builtin_prefetch(Arow + k + 64, 0, 1);   // global_prefetch_b8
    #pragma unroll
    for (int t = 0; t < 4; ++t) {
      v2f bf = *(const v2f*)(Wrow + (size_t)(t << 4) * Kd + k);
      acc[t] = __builtin_amdgcn_wmma_f32_16x16x4_f32(
          false, a, false, bf, (short)0, acc[t], false, false);
    }
  }

  // C/D layout: VGPR r -> row m0 + r + 8*half, col n0 + t*16 + l15
  const int rbase = m0 + (half << 3);
  #pragma unroll
  for (int t = 0; t < 4; ++t) {
    const int col = n0 + (t << 4) + l15;
    const float bv = bias ? bias[col] : 0.0f;
    #pragma unroll
    for (int r = 0; r < 8; ++r) {
      float val = acc[t][r] + bv;
      if (act == 1) val = 1.0f / (1.0f + __expf(-val));   // sigmoid
      const int row = rbase + r;
      if (col < splitCol)
        C0[(size_t)row * splitCol + col] = val;
      else
        C1[(size_t)row * (Nout - splitCol) + (col - splitCol)] = val;
    }
  }
}

// ---------------------------------------------------------------------------
// Depthwise causal conv (K=4) + bias:  u[b,l,d] = cb[d] + sum_k v[b,l-3+k,d]*cw[d,k]
// ---------------------------------------------------------------------------
__global__ __launch_bounds__(256) void conv_bias_k(
    const float* __restrict__ v, const float* __restrict__ cw,
    const float* __restrict__ cb, float* __restrict__ u)
{
  const int tid = blockIdx.x * blockDim.x + threadIdx.x;   // (b*L+l)*D + d
  const int d  = tid & (Dd - 1);
  const int l  = (tid >> 8) & (Ll - 1);
  float acc = cb[d];
  #pragma unroll
  for (int k = 0; k < 4; ++k) {
    const int ls = l - 3 + k;
    if (ls >= 0) acc = fmaf(v[tid + (k - 3) * Dd], cw[(d << 2) + k], acc);
  }
  u[tid] = acc;
}

// ---------------------------------------------------------------------------
// alpha2[n,d] = -8 * softplus(A[n,d]) * log2(e)   (so scan can use exp2)
// ---------------------------------------------------------------------------
__global__ __launch_bounds__(256) void alpha_k(const float* __restrict__ A,
                                               float* __restrict__ alpha)
{
  const int tid = blockIdx.x * blockDim.x + threadIdx.x;
  if (tid < Nn * Dd)
    alpha[tid] = -8.0f * log1pf(__expf(A[tid])) * 1.4426950408889634f;
}

// ---------------------------------------------------------------------------
// Pass A: per-chunk local scan. Block = (b, chunk j); thread = d.
// bc tile (CLEN x 2N = 32 KB) staged in LDS once, broadcast-read per step.
// 32 n-states per half to bound VGPRs (h+P+alpha = 96 regs).
// ---------------------------------------------------------------------------
__global__ __launch_bounds__(256) void scan_partial(
    const float* __restrict__ sg, const float* __restrict__ ug,
    const float* __restrict__ bcg, const float* __restrict__ alpha,
    float* __restrict__ hend, float* __restrict__ pend)
{
  __shared__ float lbc[CLEN * 2 * Nn];   // 32 KB
  const int d = threadIdx.x;
  const int j = blockIdx.x % NCH;
  const int b = blockIdx.x / NCH;
  const int row0 = b * Ll + j * CLEN;

  {
    const float4* src = (const float4*)(bcg + (size_t)row0 * (2 * Nn));
    float4* dst = (float4*)lbc;
    #pragma unroll
    for (int t = 0; t < (CLEN * 2 * Nn) / (4 * 256); ++t)
      dst[threadIdx.x + t * 256] = src[threadIdx.x + t * 256];
  }
  __syncthreads();

  #pragma unroll 1
  for (int nh = 0; nh < 2; ++nh) {
    const int nb = nh << 5;
    float h[32], P[32], al[32];
    #pragma unroll
    for (int i = 0; i < 32; ++i) {
      h[i] = 0.0f; P[i] = 1.0f;
      al[i] = alpha[(size_t)(nb + i) * Dd + d];
    }
    for (int l = 0; l < CLEN; ++l) {
      const int ro = row0 + l;
      const float sv = sg[(size_t)ro * Dd + d];
      const float uv = ug[(size_t)ro * Dd + d];
      const float* brow = lbc + l * (2 * Nn) + nb;   // LDS broadcast
      #pragma unroll
      for (int i = 0; i < 32; ++i) {
        const float lam = exp2f(sv * al[i]);         // v_exp_f32
        const float wsc = sqrtf(1.0f - lam * lam + 1e-6f);
        h[i] = fmaf(lam, h[i], brow[i] * uv * wsc);
        P[i] *= lam;
      }
    }
    const size_t ob = ((size_t)(b * NCH + j) * Nn + nb) * Dd + d;
    #pragma unroll
    for (int i = 0; i < 32; ++i) {
      hend[ob + (size_t)i * Dd] = h[i];
      pend[ob + (size_t)i * Dd] = P[i];
    }
  }
}

// ---------------------------------------------------------------------------
// Pass B: sequential cross-chunk combine. Thread = (b,n,d).
// hstart[j] = carry into chunk j;  H = hend[j] + pend[j]*H
// ---------------------------------------------------------------------------
__global__ __launch_bounds__(256) void scan_combine(
    const float* __restrict__ hend, const float* __restrict__ pend,
    float* __restrict__ hstart)
{
  const int tid = blockIdx.x * blockDim.x + threadIdx.x;  // over B*N*D
  const int d = tid & (Dd - 1);
  const int n = (tid >> 8) & (Nn - 1);
  const int b = tid >> 14;
  float H = 0.0f;
  for (int j = 0; j < NCH; ++j) {
    const size_t o = ((size_t)(b * NCH + j) * Nn + n) * Dd + d;
    hstart[o] = H;
    H = fmaf(pend[o], H, hend[o]);
  }
}

// ---------------------------------------------------------------------------
// Pass C: seeded chunk scan + y = sum_n c*h, fused with g = gelu(skip)*y.
// Same LDS staging of the bc tile.
// ---------------------------------------------------------------------------
__global__ __launch_bounds__(256) void scan_final(
    const float* __restrict__ sg, const float* __restrict__ ug,
    const float* __restrict__ bcg, const float* __restrict__ alpha,
    const float* __restrict__ hstart, const float* __restrict__ skip,
    float* __restrict__ g)
{
  __shared__ float lbc[CLEN * 2 * Nn];   // 32 KB
  const int d = threadIdx.x;
  const int j = blockIdx.x % NCH;
  const int b = blockIdx.x / NCH;
  const int row0 = b * Ll + j * CLEN;

  {
    const float4* src = (const float4*)(bcg + (size_t)row0 * (2 * Nn));
    float4* dst = (float4*)lbc;
    #pragma unroll
    for (int t = 0; t < (CLEN * 2 * Nn) / (4 * 256); ++t)
      dst[threadIdx.x + t * 256] = src[threadIdx.x + t * 256];
  }
  __syncthreads();

  #pragma unroll 1
  for (int nh = 0; nh < 2; ++nh) {
    const int nb = nh << 5;
    float h[32], al[32];
    const size_t ib = ((size_t)(b * NCH + j) * Nn + nb) * Dd + d;
    #pragma unroll
    for (int i = 0; i < 32; ++i) {
      h[i]  = hstart[ib + (size_t)i * Dd];
      al[i] = alpha[(size_t)(nb + i) * Dd + d];
    }
    for (int l = 0; l < CLEN; ++l) {
      const int ro = row0 + l;
      const size_t o = (size_t)ro * Dd + d;
      const float sv = sg[o];
      const float uv = ug[o];
      const float* brow = lbc + l * (2 * Nn);        // LDS broadcast
      float y = 0.0f;
      #pragma unroll
      for (int i = 0; i < 32; ++i) {
        const float lam = exp2f(sv * al[i]);
        const float wsc = sqrtf(1.0f - lam * lam + 1e-6f);
        h[i] = fmaf(lam, h[i], brow[nb + i] * uv * wsc);
        y = fmaf(brow[Nn + nb + i], h[i], y);        // c * h
      }
      if (nh == 0) {
        g[o] = y;                                    // partial over first 32 n
      } else {
        const float sk = skip[o];
        const float ge = 0.5f * sk * (1.0f + erff(sk * 0.70710678118654752f));
        g[o] = ge * (g[o] + y);                      // gelu(skip) * y
      }
    }
  }
}

// ---------------------------------------------------------------------------
extern "C" void kernel_launch(void* const* d_in, const int* in_sizes, int n_in,
                              void* d_out, int out_size, void* d_ws, size_t ws_size,
                              hipStream_t stream)
{
  (void)in_sizes; (void)n_in; (void)out_size; (void)ws_size;
  const float* x      = (const float*)d_in[0];
  const float* W_in   = (const float*)d_in[1];
  const float* conv_w = (const float*)d_in[2];
  const float* conv_b = (const float*)d_in[3];
  const float* W_bc   = (const float*)d_in[4];
  const float* b_bc   = (const float*)d_in[5];
  const float* W_lam  = (const float*)d_in[6];
  const float* b_lam  = (const float*)d_in[7];
  const float* Af     = (const float*)d_in[8];
  const float* W_out  = (const float*)d_in[9];
  float* out = (float*)d_out;

  char* w = (char*)d_ws;
  const size_t MB = (size_t)1 << 20;
  float* skip = (float*)(w + 0 * MB);    // 4 MB  (B,L,D)
  float* vbuf = (float*)(w + 4 * MB);    // 4 MB  pre-conv u; reused as g
  float* ubuf = (float*)(w + 8 * MB);    // 4 MB  (B,L,D)
  float* bcb  = (float*)(w + 12 * MB);   // 2 MB  (B,L,2N)
  float* sbuf = (float*)(w + 14 * MB);   // 4 MB  sigmoid(u@W_lam.T+b)
  float* alp  = (float*)(w + 18 * MB);   // 64 KB (N,D), pre-scaled by log2e
  float* hend = (float*)(w + 19 * MB);   // 4 MB  (B,NCH,N,D)
  float* pend = (float*)(w + 23 * MB);   // 4 MB
  float* hst  = (float*)(w + 27 * MB);   // 4 MB  -> 31 MB total
  float* g = vbuf;

  // proj = x @ W_in^T  -> skip (cols<256) | vbuf (cols>=256)
  wmma_gemm_f32<<<256, 256, 0, stream>>>(x, W_in, nullptr, skip, vbuf,
                                         Dd, Mm, Dd, 2 * Dd, 0);
  // depthwise causal conv + bias
  conv_bias_k<<<(Mm * Dd) / 256, 256, 0, stream>>>(vbuf, conv_w, conv_b, ubuf);
  // bc = u @ W_bc^T + b_bc
  wmma_gemm_f32<<<64, 256, 0, stream>>>(ubuf, W_bc, b_bc, bcb, nullptr,
                                        2 * Nn, Mm, Dd, 2 * Nn, 0);
  // s = sigmoid(u @ W_lam^T + b_lam)
  wmma_gemm_f32<<<128, 256, 0, stream>>>(ubuf, W_lam, b_lam, sbuf, nullptr,
                                         Dd, Mm, Dd, Dd, 1);
  // alpha = -8*softplus(A)*log2(e)
  alpha_k<<<(Nn * Dd) / 256, 256, 0, stream>>>(Af, alp);
  // chunked scan
  scan_partial<<<Bb * NCH, 256, 0, stream>>>(sbuf, ubuf, bcb, alp, hend, pend);
  scan_combine<<<(Bb * Nn * Dd) / 256, 256, 0, stream>>>(hend, pend, hst);
  scan_final<<<Bb * NCH, 256, 0, stream>>>(sbuf, ubuf, bcb, alp, hst, skip, g);
  // out = g @ W_out^T
  wmma_gemm_f32<<<128, 256, 0, stream>>>(g, W_out, nullptr, out, nullptr,
                                         Dd, Mm, Dd, Dd, 0);
}